// InheritanceHandler_17428977287203
// MI455X (gfx1250) — compile-verified
//
#include <hip/hip_runtime.h>

// ---------------------------------------------------------------------------
// Problem geometry (fixed by the reference).
// ---------------------------------------------------------------------------
namespace {
constexpr int B  = 2;
constexpr int C  = 32;
constexpr int PD = 32, PH = 64, PW = 64;     // coarse ("p") volume
constexpr int CD = 64, CH = 128, CW = 128;   // fine   ("c") grid

constexpr long long NFEAT = (long long)B * CD * CH * CW * C;  // 67,108,864 f32
constexpr long long NC    = (long long)B * CD * CH * CW;      //  2,097,152 points
constexpr long long NC3   = NC * 3;                           //  6,291,456 f32
constexpr long long PVOL3 = (long long)PD * PH * PW * 3;      // per-batch p_coord

constexpr int THREADS       = 256;                 // 8 wave32 per block
constexpr int INTERP_BLOCKS = (int)(NC / THREADS); // 8192 (exact)
constexpr int COPY_BLOCKS   = 8192;
constexpr long long COPY_T  = (long long)COPY_BLOCKS * THREADS; // 2,097,152 threads
constexpr int COPY_K        = 8;                   // NFEAT/4 / COPY_T (exact)

typedef float v4f __attribute__((ext_vector_type(4)));
} // namespace

// ---------------------------------------------------------------------------
// Fused kernel.
//  blocks [0, INTERP_BLOCKS)                : trilinear coordinate inheritance
//  blocks [INTERP_BLOCKS, +COPY_BLOCKS)     : 268 MB NT stream copy of c_feat
// The copy side is 8-deep software-pipelined: 8 non-temporal B128 loads in
// flight per wave, then 8 NT stores (in-order load returns => descending
// s_wait_loadcnt). The interp side gathers from the 3 MB, L2-resident p_coord
// volume; NT hints on the big stream keep it from evicting that volume.
// ---------------------------------------------------------------------------
__global__ void __launch_bounds__(THREADS)
fused_kernel(const float* __restrict__ c_feat,
             const float* __restrict__ c_coord,
             const float* __restrict__ p_coord,
             const float* __restrict__ p_start,
             float* __restrict__ out_feat,
             float* __restrict__ out_a,
             float* __restrict__ out_b) {
  // ---------------- stream-copy role ----------------
  if (blockIdx.x >= (unsigned)INTERP_BLOCKS) {
    const long long cid =
        (long long)(blockIdx.x - INTERP_BLOCKS) * THREADS + threadIdx.x;
    const v4f* s = (const v4f*)c_feat;
    v4f*       d = (v4f*)out_feat;
    v4f v[COPY_K];
#pragma unroll
    for (int k = 0; k < COPY_K; ++k)
      v[k] = __builtin_nontemporal_load(s + cid + (long long)k * COPY_T);
#pragma unroll
    for (int k = 0; k < COPY_K; ++k)
      __builtin_nontemporal_store(v[k], d + cid + (long long)k * COPY_T);
    return;
  }

  // ---------------- interpolation role ----------------
  const long long i = (long long)blockIdx.x * THREADS + threadIdx.x; // < NC
  const int b = (int)(i >> 20);            // CD*CH*CW = 2^20 points per batch

  // Fine-grid coordinate, last-dim order (z, y, x).
  const float* cc = c_coord + i * 3;
  const float gz = cc[0], gy = cc[1], gx = cc[2];

  // scaling = -1 / p_start_coord[b,0,0,0,:]
  const float* ps = p_start + (long long)b * PVOL3;
  const float sz = -1.0f / ps[0];
  const float sy = -1.0f / ps[1];
  const float sx = -1.0f / ps[2];

  // unnormalize + border clip (align_corners=True)
  float x = fminf(fmaxf((gx * sx + 1.0f) * 0.5f * (float)(PW - 1), 0.0f), (float)(PW - 1));
  float y = fminf(fmaxf((gy * sy + 1.0f) * 0.5f * (float)(PH - 1), 0.0f), (float)(PH - 1));
  float z = fminf(fmaxf((gz * sz + 1.0f) * 0.5f * (float)(PD - 1), 0.0f), (float)(PD - 1));

  const float xf = floorf(x), yf = floorf(y), zf = floorf(z);
  const float wx = x - xf,    wy = y - yf,    wz = z - zf;
  const int x0 = (int)xf, y0 = (int)yf, z0 = (int)zf;
  const int x1 = min(x0 + 1, PW - 1);
  const int y1 = min(y0 + 1, PH - 1);
  const int z1 = min(z0 + 1, PD - 1);

  const float* pb = p_coord + (long long)b * PVOL3;

  float r0 = 0.0f, r1 = 0.0f, r2 = 0.0f;
  auto acc = [&](int zi, int yi, int xi, float w) {
    const float* q = pb + ((long long)(zi * PH + yi) * PW + xi) * 3;
    r0 += (q[0] * sz) * w;   // vol channel = p_coord * s  (matches reference)
    r1 += (q[1] * sy) * w;
    r2 += (q[2] * sx) * w;
  };

  const float iwx = 1.0f - wx, iwy = 1.0f - wy, iwz = 1.0f - wz;
  // Reference accumulation order:
  acc(z0, y0, x0, iwz * iwy * iwx);
  acc(z0, y0, x1, iwz * iwy * wx);
  acc(z0, y1, x0, iwz * wy  * iwx);
  acc(z0, y1, x1, iwz * wy  * wx);
  acc(z1, y0, x0, wz  * iwy * iwx);
  acc(z1, y0, x1, wz  * iwy * wx);
  acc(z1, y1, x0, wz  * wy  * iwx);
  acc(z1, y1, x1, wz  * wy  * wx);

  const float o0 = r0 / sz;   // final "/ s" from the reference
  const float o1 = r1 / sy;
  const float o2 = r2 / sx;

  float* da = out_a + i * 3;  // 12 B/lane contiguous -> global_store_b96
  da[0] = o0; da[1] = o1; da[2] = o2;
  float* db = out_b + i * 3;
  db[0] = o0; db[1] = o1; db[2] = o2;
}

// ---------------------------------------------------------------------------
// Host-side launch. Inputs (setup_inputs order):
//   d_in[0]=c_feat  d_in[1]=c_coord  d_in[2]=p_feat(unused)
//   d_in[3]=p_coord d_in[4]=p_start_coord
// Output tuple flat: [c_feat | c_coord_inh | c_coord_inh]
// ---------------------------------------------------------------------------
extern "C" void kernel_launch(void* const* d_in, const int* in_sizes, int n_in,
                              void* d_out, int out_size, void* d_ws, size_t ws_size,
                              hipStream_t stream) {
  (void)in_sizes; (void)n_in; (void)out_size; (void)d_ws; (void)ws_size;

  const float* c_feat  = (const float*)d_in[0];
  const float* c_coord = (const float*)d_in[1];
  const float* p_coord = (const float*)d_in[3];
  const float* p_start = (const float*)d_in[4];

  float* out       = (float*)d_out;
  float* out_feat  = out;                 // 67,108,864 floats
  float* out_inh_a = out + NFEAT;         //  6,291,456 floats
  float* out_inh_b = out_inh_a + NC3;     //  6,291,456 floats

  const int blocks = INTERP_BLOCKS + COPY_BLOCKS;   // 16384
  fused_kernel<<<blocks, THREADS, 0, stream>>>(
      c_feat, c_coord, p_coord, p_start, out_feat, out_inh_a, out_inh_b);
}